// GIN_3599182594528
// MI455X (gfx1250) — compile-verified
//
#include <hip/hip_runtime.h>

typedef float v2f __attribute__((ext_vector_type(2)));
typedef float v8f __attribute__((ext_vector_type(8)));

#define N_NODES  40000
#define N_EDGES  640000
#define HDIM     128
#define F_INDIM  16
#define OUT_DIM  8
#define NGRAPH   64
#define BN_EPS   1e-5f

__device__ __forceinline__ void atomAddF(float* p, float v) {
    unsafeAtomicAdd(p, v);   // lowers to global_atomic_add_f32 on AMD
}

// ---------------------------------------------------------------- zero
__global__ void zero_kernel(float* __restrict__ p, int n) {
    for (int i = blockIdx.x * blockDim.x + threadIdx.x; i < n;
         i += gridDim.x * blockDim.x)
        p[i] = 0.0f;
}

// ---------------------------------------------------------------- edges
// msg = relu(x[src] + edge_attr @ We + be) ; aggr[dst] += msg
template <int D>
__global__ __launch_bounds__(256)
void edge_kernel(const float* __restrict__ x,
                 const int*   __restrict__ src,
                 const int*   __restrict__ dst,
                 const float* __restrict__ ea,    // [E,2]
                 const float* __restrict__ We,    // [2,D]
                 const float* __restrict__ be,    // [D]
                 float*       __restrict__ aggr)  // [N,D]
{
    constexpr int CPE = D / 4;                 // threads per edge
    int tid = blockIdx.x * blockDim.x + threadIdx.x;
    if (tid >= N_EDGES * CPE) return;
    int e  = tid / CPE;
    int f0 = (tid % CPE) * 4;

    int s = src[e], d = dst[e];
    float ea0 = ea[2 * e], ea1 = ea[2 * e + 1];

    float4 xv = *(const float4*)(x + (size_t)s * D + f0);
    float4 w0 = *(const float4*)(We + f0);
    float4 w1 = *(const float4*)(We + D + f0);
    float4 bb = *(const float4*)(be + f0);

    float m0 = fmaxf(xv.x + ea0 * w0.x + ea1 * w1.x + bb.x, 0.0f);
    float m1 = fmaxf(xv.y + ea0 * w0.y + ea1 * w1.y + bb.y, 0.0f);
    float m2 = fmaxf(xv.z + ea0 * w0.z + ea1 * w1.z + bb.z, 0.0f);
    float m3 = fmaxf(xv.w + ea0 * w0.w + ea1 * w1.w + bb.w, 0.0f);

    float* ap = aggr + (size_t)d * D + f0;
    atomAddF(ap + 0, m0);
    atomAddF(ap + 1, m1);
    atomAddF(ap + 2, m2);
    atomAddF(ap + 3, m3);
}

// ---------------------------------------------------------------- GEMM (WMMA f32 16x16x4)
// out[N,128] = A @ W + bias ; A is either (A0 + A1) or relu(A0*scale+shift)
// 256 threads = 8 waves; wave w: row-tile (w&1)*16, col-tile (w>>1)*32 (2 WMMA n-tiles)
#define LDS_PITCH 144   // 2*144 = 288 ≡ 32 (mod 64 banks) -> conflict-free B reads

template <int K, bool BN_IN, bool RELU_OUT>
__global__ __launch_bounds__(256)
void gemm_kernel(const float* __restrict__ A0,     // [N,K]
                 const float* __restrict__ A1,     // [N,K] (aggr) or unused
                 const float* __restrict__ scale,  // [K]  (BN) or unused
                 const float* __restrict__ shift,  // [K]  (BN) or unused
                 const float* __restrict__ W,      // [K,128]
                 const float* __restrict__ bias,   // [128]
                 float*       __restrict__ out)    // [N,128]
{
    __shared__ float sW[K * LDS_PITCH];
    __shared__ float sScale[HDIM];
    __shared__ float sShift[HDIM];

    int t = threadIdx.x;
    for (int i = t; i < K * 128; i += 256)
        sW[(i >> 7) * LDS_PITCH + (i & 127)] = W[i];
    if (BN_IN && t < HDIM) { sScale[t] = scale[t]; sShift[t] = shift[t]; }
    __syncthreads();

    int w     = t >> 5;
    int l     = t & 31;
    int ln16  = l & 15;
    int half  = l >> 4;          // 0: lanes 0-15, 1: lanes 16-31
    int rtile = w & 1;
    int n0    = (w >> 1) * 32;

    int row  = blockIdx.x * 32 + rtile * 16 + ln16;  // A-fragment row (M = lane&15)
    int koff = half * 2;                             // A/B K offset per half-wave

    const float* arow0 = A0 + (size_t)row * K;
    const float* arow1 = BN_IN ? nullptr : (A1 + (size_t)row * K);

    v8f acc0 = {};
    v8f acc1 = {};

#pragma unroll 8
    for (int kb = 0; kb < K; kb += 4) {
        int k = kb + koff;
        float x0 = arow0[k];
        float x1 = arow0[k + 1];
        if (BN_IN) {
            x0 = fmaxf(x0 * sScale[k]     + sShift[k],     0.0f);
            x1 = fmaxf(x1 * sScale[k + 1] + sShift[k + 1], 0.0f);
        } else {
            x0 += arow1[k];
            x1 += arow1[k + 1];
        }
        v2f a; a.x = x0; a.y = x1;

        v2f b0, b1;
        b0.x = sW[k       * LDS_PITCH + n0 + ln16];
        b0.y = sW[(k + 1) * LDS_PITCH + n0 + ln16];
        b1.x = sW[k       * LDS_PITCH + n0 + 16 + ln16];
        b1.y = sW[(k + 1) * LDS_PITCH + n0 + 16 + ln16];

        acc0 = __builtin_amdgcn_wmma_f32_16x16x4_f32(false, a, false, b0,
                                                     (short)0, acc0, false, false);
        acc1 = __builtin_amdgcn_wmma_f32_16x16x4_f32(false, a, false, b1,
                                                     (short)0, acc1, false, false);
    }

    // D layout: vgpr r -> M = r + 8*half ; N = lane&15
    int mbase = blockIdx.x * 32 + rtile * 16 + half * 8;
    int col0  = n0 + ln16;
    int col1  = col0 + 16;
    float bb0 = bias[col0];
    float bb1 = bias[col1];
#pragma unroll
    for (int r = 0; r < 8; ++r) {
        float v0 = acc0[r] + bb0;
        float v1 = acc1[r] + bb1;
        if (RELU_OUT) { v0 = fmaxf(v0, 0.0f); v1 = fmaxf(v1, 0.0f); }
        out[(size_t)(mbase + r) * HDIM + col0] = v0;
        out[(size_t)(mbase + r) * HDIM + col1] = v1;
    }
}

// ---------------------------------------------------------------- BN stats
__global__ __launch_bounds__(128)
void bnstat_kernel(const float* __restrict__ z, float* __restrict__ sums) {
    int c = threadIdx.x;                         // column 0..127
    float s = 0.0f, ss = 0.0f;
    for (int r = blockIdx.x; r < N_NODES; r += gridDim.x) {
        float v = z[(size_t)r * HDIM + c];
        s += v; ss += v * v;
    }
    atomAddF(&sums[c], s);
    atomAddF(&sums[HDIM + c], ss);
}

__global__ __launch_bounds__(128)
void bnfin_kernel(const float* __restrict__ sums,
                  const float* __restrict__ g, const float* __restrict__ bt,
                  float* __restrict__ scale, float* __restrict__ shift) {
    int c = threadIdx.x;
    float mean = sums[c] * (1.0f / N_NODES);
    float var  = sums[HDIM + c] * (1.0f / N_NODES) - mean * mean;
    float sc   = g[c] * rsqrtf(var + BN_EPS);
    scale[c] = sc;
    shift[c] = bt[c] - mean * sc;
}

// ---------------------------------------------------------------- pooling
__global__ __launch_bounds__(256)
void pool_kernel(const float* __restrict__ x, const int* __restrict__ batch,
                 float* __restrict__ pooled) {
    int tid = blockIdx.x * blockDim.x + threadIdx.x;
    if (tid >= N_NODES * 32) return;
    int node = tid >> 5;
    int c4   = (tid & 31) * 4;
    int b    = batch[node];
    float4 v = *(const float4*)&x[(size_t)node * HDIM + c4];
    float* p = &pooled[b * HDIM + c4];
    atomAddF(p + 0, v.x);
    atomAddF(p + 1, v.y);
    atomAddF(p + 2, v.z);
    atomAddF(p + 3, v.w);
}

// ---------------------------------------------------------------- final MLP (tiny)
__global__ __launch_bounds__(128)
void mlp_kernel(const float* __restrict__ pooled,
                const float* __restrict__ Wf1, const float* __restrict__ bf1,
                const float* __restrict__ Wf2, const float* __restrict__ bf2,
                float* __restrict__ out) {
    __shared__ float ps[HDIM];
    __shared__ float hs[HDIM];
    int g = blockIdx.x;
    int t = threadIdx.x;
    ps[t] = pooled[g * HDIM + t];
    __syncthreads();
    float acc = bf1[t];
#pragma unroll 8
    for (int k = 0; k < HDIM; ++k) acc += ps[k] * Wf1[k * HDIM + t];
    hs[t] = fmaxf(acc, 0.0f);
    __syncthreads();
    if (t < OUT_DIM) {
        float o = bf2[t];
#pragma unroll 8
        for (int k = 0; k < HDIM; ++k) o += hs[k] * Wf2[k * OUT_DIM + t];
        out[g * OUT_DIM + t] = o;
    }
}

// ---------------------------------------------------------------- launch
extern "C" void kernel_launch(void* const* d_in, const int* in_sizes, int n_in,
                              void* d_out, int out_size, void* d_ws, size_t ws_size,
                              hipStream_t stream) {
    const float* x_in  = (const float*)d_in[0];
    const int*   ei    = (const int*)  d_in[1];
    const float* ea    = (const float*)d_in[2];
    const int*   batch = (const int*)  d_in[3];
    const int* src = ei;
    const int* dst = ei + N_EDGES;
    auto P = [&](int i) { return (const float*)d_in[i]; };

    float* ws      = (float*)d_ws;
    float* xA      = ws;
    float* xB      = xA + (size_t)N_NODES * HDIM;
    float* aggr    = xB + (size_t)N_NODES * HDIM;
    float* z1      = aggr + (size_t)N_NODES * HDIM;
    float* sums    = z1 + (size_t)N_NODES * HDIM;   // 256 floats
    float* bnscale = sums + 2 * HDIM;               // 128
    float* bnshift = bnscale + HDIM;                // 128
    float* pooled  = bnshift + HDIM;                // 64*128

    // ---------------- layer 0 (d_in = 16) ----------------
    {
        int base = 5;
        const float *We = P(base + 0), *be = P(base + 1);
        const float *W1 = P(base + 2), *b1 = P(base + 3);
        const float *g  = P(base + 4), *bt = P(base + 5);
        const float *W2 = P(base + 6), *b2 = P(base + 7);

        zero_kernel<<<1024, 256, 0, stream>>>(aggr, N_NODES * F_INDIM);
        int th = N_EDGES * (F_INDIM / 4);
        edge_kernel<F_INDIM><<<(th + 255) / 256, 256, 0, stream>>>(
            x_in, src, dst, ea, We, be, aggr);
        gemm_kernel<F_INDIM, false, false><<<N_NODES / 32, 256, 0, stream>>>(
            x_in, aggr, nullptr, nullptr, W1, b1, z1);
        zero_kernel<<<1, 256, 0, stream>>>(sums, 2 * HDIM);
        bnstat_kernel<<<512, 128, 0, stream>>>(z1, sums);
        bnfin_kernel<<<1, 128, 0, stream>>>(sums, g, bt, bnscale, bnshift);
        gemm_kernel<HDIM, true, true><<<N_NODES / 32, 256, 0, stream>>>(
            z1, nullptr, bnscale, bnshift, W2, b2, xA);
    }

    // ---------------- layers 1, 2 (d = 128) ----------------
    const float* xin = xA;
    float*       xout = xB;
    for (int lyr = 1; lyr <= 2; ++lyr) {
        int base = 5 + 8 * lyr;
        const float *We = P(base + 0), *be = P(base + 1);
        const float *W1 = P(base + 2), *b1 = P(base + 3);
        const float *g  = P(base + 4), *bt = P(base + 5);
        const float *W2 = P(base + 6), *b2 = P(base + 7);

        zero_kernel<<<2048, 256, 0, stream>>>(aggr, N_NODES * HDIM);
        int th = N_EDGES * (HDIM / 4);
        edge_kernel<HDIM><<<(th + 255) / 256, 256, 0, stream>>>(
            xin, src, dst, ea, We, be, aggr);
        gemm_kernel<HDIM, false, false><<<N_NODES / 32, 256, 0, stream>>>(
            xin, aggr, nullptr, nullptr, W1, b1, z1);
        zero_kernel<<<1, 256, 0, stream>>>(sums, 2 * HDIM);
        bnstat_kernel<<<512, 128, 0, stream>>>(z1, sums);
        bnfin_kernel<<<1, 128, 0, stream>>>(sums, g, bt, bnscale, bnshift);
        gemm_kernel<HDIM, true, true><<<N_NODES / 32, 256, 0, stream>>>(
            z1, nullptr, bnscale, bnshift, W2, b2, xout);

        const float* tmp = xin;
        xin  = xout;
        xout = (float*)tmp;
    }
    // after layer1: xA->xB ; after layer2: xB->xA  => xin == xA

    // ---------------- pooling + final MLP ----------------
    zero_kernel<<<32, 256, 0, stream>>>(pooled, NGRAPH * HDIM);
    pool_kernel<<<(N_NODES * 32 + 255) / 256, 256, 0, stream>>>(xin, batch, pooled);
    mlp_kernel<<<NGRAPH, HDIM, 0, stream>>>(pooled, P(29), P(30), P(31), P(32),
                                            (float*)d_out);
}